// SocialItemEmbedding_9216999817726
// MI455X (gfx1250) — compile-verified
//
#include <hip/hip_runtime.h>

// SocialItemEmbedding: 2-layer GATv2 pipeline for MI455X (gfx1250, wave32).
// GEMMs use V_WMMA_F32_16X16X4_F32 (fp32-exact); edge phase is gather-bound
// wave-per-edge float4 traffic (lane*4 == 128 floats on wave32).

typedef __attribute__((ext_vector_type(2))) float v2f;
typedef __attribute__((ext_vector_type(8))) float v8f;

#define FDIM 128

__device__ __forceinline__ float lrelu(float x, float s) { return x > 0.0f ? x : x * s; }

// monotone float max via int atomics (init bits must be 0xFF800000 == -inf)
__device__ __forceinline__ void atomicMaxFloat(float* addr, float val) {
  if (val >= 0.0f) atomicMax((int*)addr, __float_as_int(val));
  else             atomicMin((unsigned int*)addr, (unsigned int)__float_as_int(val));
}

__global__ void fill_kernel(float* __restrict__ p, float v, long n) {
  long i = (long)blockIdx.x * blockDim.x + threadIdx.x;
  if (i < n) p[i] = v;
}

__global__ void copy_kernel(float* __restrict__ dst, const float* __restrict__ src, long n) {
  long i = (long)blockIdx.x * blockDim.x + threadIdx.x;
  if (i < n) dst[i] = src[i];
}

__global__ void leaky_inplace_kernel(float* __restrict__ p, float slope, long n) {
  long i = (long)blockIdx.x * blockDim.x + threadIdx.x;
  if (i < n) p[i] = lrelu(p[i], slope);
}

// Y[r,:] = X[rowIdx? rowIdx[r] : r, :] @ W + bias.  F=128 fixed.
// CONTRACT: Nrows % 16 == 0 (true here: 50000, 100000), so no tail guards and
// EXEC stays all-ones through every WMMA (ISA requirement).
// Block = 256 threads = 8 waves; wave w -> 16x16 tile (rows blk*16.., cols w*16..),
// K loop = 32 x V_WMMA_F32_16X16X4_F32.
// ISA layouts (32-bit): A 16x4: v0 = {K0 | K2}, v1 = {K1 | K3} split at lane16.
// B 4x16: same K split, N = lane&15. D: vgpr i = rows {i | i+8}, N = lane&15.
__global__ void gemm128_bias_wmma(const float* __restrict__ X,
                                  const int* __restrict__ rowIdx,
                                  const float* __restrict__ W,
                                  const float* __restrict__ bias,
                                  float* __restrict__ Y, int Nrows) {
  const int lane  = threadIdx.x & 31;
  const int wave  = threadIdx.x >> 5;
  const int row0  = blockIdx.x << 4;
  const int col0  = wave << 4;
  const int m     = lane & 15;
  const int g     = lane >> 4;     // half-wave select
  const int khalf = g << 1;        // K offset 0 or 2
  const int nc    = col0 + m;      // output column for this lane
  const int r     = row0 + m;      // A row for this lane
  const long xr   = rowIdx ? (long)rowIdx[r] : (long)r;
  // A-fragment: the two K values per lane are adjacent -> one b64 load / iter
  const float2* xp = (const float2*)(X + xr * FDIM);

  v8f acc = {};
#pragma unroll
  for (int k0 = 0; k0 < FDIM; k0 += 4) {
    const float2 av = xp[(k0 >> 1) + g];
    v2f a, b;
    a.x = av.x;
    a.y = av.y;
    b.x = W[(k0 + khalf)     * FDIM + nc];
    b.y = W[(k0 + khalf + 1) * FDIM + nc];
    acc = __builtin_amdgcn_wmma_f32_16x16x4_f32(false, a, false, b, (short)0,
                                                acc, false, false);
  }
  const float bv = bias[nc];
  const int  mhi = g << 3;         // D rows i / i+8
  float* yp = Y + (long)(row0 + mhi) * FDIM + nc;
#pragma unroll
  for (int i = 0; i < 8; ++i) {
    yp[(long)i * FDIM] = acc[i] + bv;
  }
}

// s[e] = leaky_relu(el[src]+er[dst], 0.2) . attn ; segment-max into m[dst].
// One wave per edge: lane loads float4 -> 32*4 = 128 features.
__global__ void edge_logits_kernel(const int* __restrict__ src, const int* __restrict__ dst,
                                   const float* __restrict__ el, const float* __restrict__ er,
                                   const float* __restrict__ attn,
                                   float* __restrict__ s, float* __restrict__ mseg, int E) {
  const int e    = blockIdx.x * 8 + (threadIdx.x >> 5);
  const int lane = threadIdx.x & 31;
  if (e >= E) return;
  const int u = src[e], v = dst[e];
  const float4 a = ((const float4*)(el + (long)u * FDIM))[lane];
  const float4 b = ((const float4*)(er + (long)v * FDIM))[lane];
  const float4 w = ((const float4*)attn)[lane];
  float p = lrelu(a.x + b.x, 0.2f) * w.x
          + lrelu(a.y + b.y, 0.2f) * w.y
          + lrelu(a.z + b.z, 0.2f) * w.z
          + lrelu(a.w + b.w, 0.2f) * w.w;
#pragma unroll
  for (int off = 16; off > 0; off >>= 1) p += __shfl_xor(p, off, 32);
  if (lane == 0) {
    s[e] = p;
    atomicMaxFloat(&mseg[v], p);
  }
}

// s[e] <- exp(s[e]-m[dst]);  den[dst] += s[e]
__global__ void edge_exp_kernel(float* __restrict__ s, const float* __restrict__ mseg,
                                float* __restrict__ den, const int* __restrict__ dst, int E) {
  const int e = blockIdx.x * blockDim.x + threadIdx.x;
  if (e >= E) return;
  const int v = dst[e];
  const float ex = __expf(s[e] - mseg[v]);
  s[e] = ex;
  atomicAdd(&den[v], ex);
}

// out[dst] += el[src] * (ex[e]/den[dst]).  Wave per edge, float atomics.
__global__ void edge_aggregate_kernel(const int* __restrict__ src, const int* __restrict__ dst,
                                      const float* __restrict__ el, const float* __restrict__ ex,
                                      const float* __restrict__ den, float* __restrict__ out, int E) {
  const int e    = blockIdx.x * 8 + (threadIdx.x >> 5);
  const int lane = threadIdx.x & 31;
  if (e >= E) return;
  const int u = src[e], v = dst[e];
  const float alpha = ex[e] / den[v];
  const float4 a = ((const float4*)(el + (long)u * FDIM))[lane];
  float* op = out + (long)v * FDIM + lane * 4;
  atomicAdd(op + 0, a.x * alpha);
  atomicAdd(op + 1, a.y * alpha);
  atomicAdd(op + 2, a.z * alpha);
  atomicAdd(op + 3, a.w * alpha);
}

// h = leaky_relu(h1acc, 0.01); if (rowsum(h) != 0) soc[nid[r]] = h
__global__ void scatter_h1_kernel(const float* __restrict__ h1, const int* __restrict__ nid,
                                  float* __restrict__ soc, int Nrows) {
  const int row  = blockIdx.x * 8 + (threadIdx.x >> 5);
  const int lane = threadIdx.x & 31;
  if (row >= Nrows) return;
  float4 h = ((const float4*)(h1 + (long)row * FDIM))[lane];
  h.x = lrelu(h.x, 0.01f); h.y = lrelu(h.y, 0.01f);
  h.z = lrelu(h.z, 0.01f); h.w = lrelu(h.w, 0.01f);
  float p = h.x + h.y + h.z + h.w;
#pragma unroll
  for (int off = 16; off > 0; off >>= 1) p += __shfl_xor(p, off, 32);
  if (p != 0.0f) {
    ((float4*)(soc + (long)nid[row] * FDIM))[lane] = h;
  }
}

extern "C" void kernel_launch(void* const* d_in, const int* in_sizes, int n_in,
                              void* d_out, int out_size, void* d_ws, size_t ws_size,
                              hipStream_t stream) {
  const float* emb     = (const float*)d_in[0];
  const int*   i2u_src = (const int*)  d_in[1];
  const int*   i2u_dst = (const int*)  d_in[2];
  const int*   i2u_nid = (const int*)  d_in[3];
  const int*   soc_src = (const int*)  d_in[4];
  const int*   soc_dst = (const int*)  d_in[5];
  const float* Wsrc1   = (const float*)d_in[6];
  const float* bsrc1   = (const float*)d_in[7];
  const float* Wdst1   = (const float*)d_in[8];
  const float* bdst1   = (const float*)d_in[9];
  const float* attn1   = (const float*)d_in[10];
  const float* Wsrc2   = (const float*)d_in[11];
  const float* bsrc2   = (const float*)d_in[12];
  const float* Wdst2   = (const float*)d_in[13];
  const float* bdst2   = (const float*)d_in[14];
  const float* attn2   = (const float*)d_in[15];

  const int E1 = in_sizes[1];          // 800000
  const int N1 = in_sizes[3];          // 50000  (multiple of 16)
  const int E2 = in_sizes[4];          // 1600000
  const int N2 = in_sizes[0] / FDIM;   // 100000 (multiple of 16)

  float* ws = (float*)d_ws;
  // workspace layout (floats); el2 reuses the el1/er1 region after soc is built
  float* el1  = ws;                          // [N1*128]
  float* er1  = ws + (long)N1 * FDIM;        // [N1*128]
  float* h1   = ws + (long)2 * N1 * FDIM;    // [N1*128]
  float* el2  = ws;                          // [N2*128]  (aliases el1+er1, used later)
  long   off  = (long)3 * N1 * FDIM;
  float* s1   = ws + off;  off += E1;        // logits/exp layer1
  float* m1   = ws + off;  off += N1;
  float* den1 = ws + off;  off += N1;
  float* soc  = ws + off;  off += (long)N2 * FDIM;
  float* er2  = ws + off;  off += (long)N2 * FDIM;
  float* s2   = ws + off;  off += E2;
  float* m2   = ws + off;  off += N2;
  float* den2 = ws + off;  off += N2;
  float* outp = (float*)d_out;
  (void)ws_size; (void)n_in; (void)out_size;

  const float NEG_INF = -__builtin_huge_valf();
  const long  feat1 = (long)N1 * FDIM, feat2 = (long)N2 * FDIM;
  auto cdiv = [](long a, long b) { return (unsigned)((a + b - 1) / b); };

  // ---- Layer 1: projections (gather rows via i2u_nid) ----
  gemm128_bias_wmma<<<cdiv(N1, 16), 256, 0, stream>>>(emb, i2u_nid, Wsrc1, bsrc1, el1, N1);
  gemm128_bias_wmma<<<cdiv(N1, 16), 256, 0, stream>>>(emb, i2u_nid, Wdst1, bdst1, er1, N1);

  fill_kernel<<<cdiv(N1, 256), 256, 0, stream>>>(m1, NEG_INF, N1);
  fill_kernel<<<cdiv(N1, 256), 256, 0, stream>>>(den1, 0.0f, N1);
  fill_kernel<<<cdiv(feat1, 256), 256, 0, stream>>>(h1, 0.0f, feat1);

  edge_logits_kernel<<<cdiv(E1, 8), 256, 0, stream>>>(i2u_src, i2u_dst, el1, er1, attn1, s1, m1, E1);
  edge_exp_kernel<<<cdiv(E1, 256), 256, 0, stream>>>(s1, m1, den1, i2u_dst, E1);
  edge_aggregate_kernel<<<cdiv(E1, 8), 256, 0, stream>>>(i2u_src, i2u_dst, el1, s1, den1, h1, E1);

  // ---- soc = embedding; soc[nid] = leaky(h1) where rowsum != 0 ----
  copy_kernel<<<cdiv(feat2, 256), 256, 0, stream>>>(soc, emb, feat2);
  scatter_h1_kernel<<<cdiv(N1, 8), 256, 0, stream>>>(h1, i2u_nid, soc, N1);

  // ---- Layer 2: projections ----
  gemm128_bias_wmma<<<cdiv(N2, 16), 256, 0, stream>>>(soc, nullptr, Wsrc2, bsrc2, el2, N2);
  gemm128_bias_wmma<<<cdiv(N2, 16), 256, 0, stream>>>(soc, nullptr, Wdst2, bdst2, er2, N2);

  fill_kernel<<<cdiv(N2, 256), 256, 0, stream>>>(m2, NEG_INF, N2);
  fill_kernel<<<cdiv(N2, 256), 256, 0, stream>>>(den2, 0.0f, N2);
  fill_kernel<<<cdiv(feat2, 256), 256, 0, stream>>>(outp, 0.0f, feat2);

  edge_logits_kernel<<<cdiv(E2, 8), 256, 0, stream>>>(soc_src, soc_dst, el2, er2, attn2, s2, m2, E2);
  edge_exp_kernel<<<cdiv(E2, 256), 256, 0, stream>>>(s2, m2, den2, soc_dst, E2);
  edge_aggregate_kernel<<<cdiv(E2, 8), 256, 0, stream>>>(soc_src, soc_dst, el2, s2, den2, outp, E2);

  leaky_inplace_kernel<<<cdiv(feat2, 256), 256, 0, stream>>>(outp, 0.01f, feat2);
}